// S4DKernel_23656679866577
// MI455X (gfx1250) — compile-verified
//
#include <hip/hip_runtime.h>

// S4D Vandermonde kernel for MI455X (gfx1250, wave32).
//
// K[h, 64*l1 + l0] = 2*Re( sum_n Ccoef[h,n] * exp(dtA[h,n]*64*l1) * exp(dtA[h,n]*l0) )
// => per-head real GEMM (64 x 64 x 64) on the WMMA f32 16x16x4 pipe.
// Contraction index k packs (n, re/im): A[l1,2n]=Re(P), A[l1,2n+1]=-Im(P),
// B[2n,l0]=Re(U), B[2n+1,l0]=Im(U).

typedef __attribute__((ext_vector_type(2))) float v2f;
typedef __attribute__((ext_vector_type(8))) float v8f;

#if defined(__gfx1250__)
#if !__has_builtin(__builtin_amdgcn_wmma_f32_16x16x4_f32)
#error "missing __builtin_amdgcn_wmma_f32_16x16x4_f32 on gfx1250"
#endif
#endif

#define H_DIM 1024
#define NHALF 32
#define L_DIM 4096   // = 64 * 64

__global__ __launch_bounds__(32)
void s4d_vandermonde_wmma(const float* __restrict__ C,
                          const float* __restrict__ log_dt,
                          const float* __restrict__ log_A_real,
                          const float* __restrict__ A_imag,
                          float* __restrict__ out)
{
    const int h    = blockIdx.x;
    const int lane = threadIdx.x & 31;
    const int half = lane >> 4;    // 0: lanes 0-15, 1: lanes 16-31
    const int mrow = lane & 15;

    const float dt = __expf(log_dt[h]);

    const float* Cb  = C          + (size_t)h * NHALF * 2;
    const float* lAb = log_A_real + (size_t)h * NHALF;
    const float* Aib = A_imag     + (size_t)h * NHALF;

    // Per-lane n for K-chunk c is n = 2c + half (matches f32 A/B VGPR layouts).
    float ar[16], ai[16];     // Re/Im of dtA for this lane's n per chunk
    v2f   afrag[4][16];       // A fragments: 4 row tiles (l1) x 16 K-chunks

#pragma unroll
    for (int c = 0; c < 16; ++c) {
        const int   n    = 2 * c + half;
        const float Are  = -__expf(lAb[n]);
        const float Aim  = Aib[n];
        const float dtar = Are * dt;
        const float dtai = Aim * dt;
        ar[c] = dtar;
        ai[c] = dtai;

        // Ccoef = 2 * Cc * (exp(dtA) - 1) / A   (complex)
        float s, co;
        const float e = __expf(dtar);
        __sincosf(dtai, &s, &co);
        const float Er  = e * co - 1.0f;
        const float Ei  = e * s;
        const float inv = 1.0f / (Are * Are + Aim * Aim);
        const float Dr  = (Er * Are + Ei * Aim) * inv;
        const float Di  = (Ei * Are - Er * Aim) * inv;
        const float Ccr = Cb[2 * n + 0];
        const float Cci = Cb[2 * n + 1];
        const float cr  = 2.0f * (Ccr * Dr - Cci * Di);
        const float ci  = 2.0f * (Ccr * Di + Cci * Dr);

        // A fragments: P = Ccoef * exp(dtA * 64*l1); store (Re(P), -Im(P))
#pragma unroll
        for (int m = 0; m < 4; ++m) {
            const float l1s = (float)((16 * m + mrow) * 64);
            const float mag = __expf(dtar * l1s);
            float ws, wc;
            __sincosf(dtai * l1s, &ws, &wc);
            const float Wr = mag * wc, Wi = mag * ws;
            v2f a;
            a.x =  cr * Wr - ci * Wi;    // k = 2n   : +Re(P)
            a.y = -(cr * Wi + ci * Wr);  // k = 2n+1 : -Im(P)
            afrag[m][c] = a;
        }
    }

    float* outb = out + (size_t)h * L_DIM;

#pragma unroll
    for (int t = 0; t < 4; ++t) {
        const float l0 = (float)(16 * t + mrow);

        // B fragments for this column tile: U = exp(dtA * l0) -> (Re, Im)
        v2f bfrag[16];
#pragma unroll
        for (int c = 0; c < 16; ++c) {
            const float mag = __expf(ar[c] * l0);
            float us, uc;
            __sincosf(ai[c] * l0, &us, &uc);
            v2f b;
            b.x = mag * uc;   // k = 2n   : Re(U)
            b.y = mag * us;   // k = 2n+1 : Im(U)
            bfrag[c] = b;
        }

        v8f acc[4] = {};
        // 4 independent accumulation chains (m) per K-step to hide WMMA latency
#pragma unroll
        for (int c = 0; c < 16; ++c) {
#pragma unroll
            for (int m = 0; m < 4; ++m) {
                acc[m] = __builtin_amdgcn_wmma_f32_16x16x4_f32(
                    /*neg_a=*/false, afrag[m][c],
                    /*neg_b=*/false, bfrag[c],
                    /*c_mod=*/(short)0, acc[m],
                    /*reuse_a=*/false, /*reuse_b=*/false);
            }
        }

        // D layout: VGPR r -> row M = r (lanes 0-15) / r+8 (lanes 16-31), N = mrow
#pragma unroll
        for (int m = 0; m < 4; ++m) {
#pragma unroll
            for (int r = 0; r < 8; ++r) {
                const int l1 = 16 * m + r + 8 * half;
                outb[l1 * 64 + 16 * t + mrow] = acc[m][r];
            }
        }
    }
}

extern "C" void kernel_launch(void* const* d_in, const int* in_sizes, int n_in,
                              void* d_out, int out_size, void* d_ws, size_t ws_size,
                              hipStream_t stream) {
    const float* C          = (const float*)d_in[0];
    const float* log_dt     = (const float*)d_in[1];
    const float* log_A_real = (const float*)d_in[2];
    const float* A_imag     = (const float*)d_in[3];
    // d_in[4] = length (int32, 0..L-1) is implicit in the kernel's indexing.
    float* out = (float*)d_out;

    dim3 grid(H_DIM);
    dim3 block(32);   // one wave32 per head
    hipLaunchKernelGGL(s4d_vandermonde_wmma, grid, block, 0, stream,
                       C, log_dt, log_A_real, A_imag, out);
}